// GraphConv_7627861917712
// MI455X (gfx1250) — compile-verified
//
#include <hip/hip_runtime.h>
#include <hip/hip_bf16.h>

typedef __attribute__((ext_vector_type(2))) float v2f;
typedef __attribute__((ext_vector_type(8))) float v8f;

#define NN     100000
#define INCH   128
#define OUTCH  128
#define NREL   4
#define NEDGE  640000

// ---------------------------------------------------------------------------
// Zero d_out (float4-vectorized). Must run every launch: graph replay does not
// re-poison, and the scatter kernel accumulates with atomics.
// ---------------------------------------------------------------------------
__global__ void gc_zero_kernel(float4* __restrict__ out, int n4) {
    int i = blockIdx.x * blockDim.x + threadIdx.x;
    if (i < n4) out[i] = make_float4(0.f, 0.f, 0.f, 0.f);
}

// ---------------------------------------------------------------------------
// att = softmax(a_att), 4 elements -> workspace
// ---------------------------------------------------------------------------
__global__ void gc_softmax_kernel(const float* __restrict__ a_att,
                                  float* __restrict__ att) {
    if (threadIdx.x == 0 && blockIdx.x == 0) {
        float m = a_att[0];
        #pragma unroll
        for (int i = 1; i < NREL; ++i) m = fmaxf(m, a_att[i]);
        float e[NREL];
        float s = 0.f;
        #pragma unroll
        for (int i = 0; i < NREL; ++i) { e[i] = __expf(a_att[i] - m); s += e[i]; }
        #pragma unroll
        for (int i = 0; i < NREL; ++i) att[i] = e[i] / s;
    }
}

// ---------------------------------------------------------------------------
// support = inputs @ w + b using V_WMMA_F32_16X16X4_F32 (full f32 precision).
// One wave computes one 16x16 output tile; 8 waves/block cover the 8 column
// tiles of OUTCH=128 for one 16-row strip. K=128 -> 32 WMMAs per tile.
//
// A fragment (16x4 f32, ISA 7.12.2): lane L holds row M=L%16,
//   VGPR j holds K = kk + (L/16)*2 + j.
// B fragment (4x16): VGPR j holds K = kk + (L/16)*2 + j, column N = L%16.
// C/D: lane L holds column N=L%16; VGPR r holds row M = r + (L/16)*8.
// ---------------------------------------------------------------------------
__global__ __launch_bounds__(256) void gc_gemm_wmma_kernel(
    const float* __restrict__ A,     // [NN, INCH]
    const float* __restrict__ W,     // [INCH, OUTCH]
    const float* __restrict__ bias,  // [OUTCH]
    float* __restrict__ S)           // [NN, OUTCH]
{
    const int lane  = threadIdx.x & 31;
    const int wv    = threadIdx.x >> 5;    // wave id 0..7 -> column tile
    const int lmod  = lane & 15;
    const int lhalf = lane >> 4;           // 0 or 1
    const int row0  = blockIdx.x << 4;     // 16-row strip
    const int col0  = wv << 4;             // 16-col tile

    const float* arow = A + (size_t)(row0 + lmod) * INCH;

    v8f acc = {};
    #pragma unroll
    for (int kk = 0; kk < INCH; kk += 4) {
        const int k = kk + (lhalf << 1);
        v2f a;
        a.x = arow[k];
        a.y = arow[k + 1];
        v2f b;
        b.x = W[(size_t)k * OUTCH + col0 + lmod];
        b.y = W[(size_t)(k + 1) * OUTCH + col0 + lmod];
        acc = __builtin_amdgcn_wmma_f32_16x16x4_f32(
            /*neg_a=*/false, a, /*neg_b=*/false, b,
            /*c_mod=*/(short)0, acc, /*reuse_a=*/false, /*reuse_b=*/false);
    }

    const float bv    = bias[col0 + lmod];
    const int   rbase = row0 + (lhalf << 3);
    #pragma unroll
    for (int r = 0; r < 8; ++r)
        S[(size_t)(rbase + r) * OUTCH + col0 + lmod] = acc[r] + bv;
}

// ---------------------------------------------------------------------------
// Sparse aggregation: one wave per (relation, edge). 32 lanes x float4 = 512
// contiguous bytes per gather (coalesced), then 4 hardware f32 atomic adds
// per lane into out[row]. grid = (NEDGE/8, NREL), 8 waves per 256-thr block.
// ---------------------------------------------------------------------------
__global__ __launch_bounds__(256) void gc_scatter_kernel(
    const float* __restrict__ support,  // [NN, OUTCH]
    const float* __restrict__ att,      // [NREL] softmaxed
    const float* __restrict__ adj_val,  // [NREL, NEDGE]
    const int*   __restrict__ row,      // [NREL, NEDGE]
    const int*   __restrict__ col,      // [NREL, NEDGE]
    float*       __restrict__ out)      // [NN, OUTCH]
{
    const int  lane = threadIdx.x & 31;
    const int  wv   = threadIdx.x >> 5;
    const long e    = (long)blockIdx.x * 8 + wv;
    const int  rel  = blockIdx.y;
    const long idx  = (long)rel * NEDGE + e;

    const int   r    = row[idx];
    const int   c    = col[idx];
    const float coef = att[rel] * adj_val[idx];

    const float4 s = *(const float4*)(support + (size_t)c * OUTCH + (lane << 2));
    float* o = out + (size_t)r * OUTCH + (lane << 2);

    unsafeAtomicAdd(o + 0, coef * s.x);
    unsafeAtomicAdd(o + 1, coef * s.y);
    unsafeAtomicAdd(o + 2, coef * s.z);
    unsafeAtomicAdd(o + 3, coef * s.w);
}

// ---------------------------------------------------------------------------
// LeakyReLU(0.2) in place over d_out
// ---------------------------------------------------------------------------
__global__ void gc_leaky_kernel(float4* __restrict__ out, int n4) {
    int i = blockIdx.x * blockDim.x + threadIdx.x;
    if (i < n4) {
        float4 v = out[i];
        v.x = v.x >= 0.f ? v.x : 0.2f * v.x;
        v.y = v.y >= 0.f ? v.y : 0.2f * v.y;
        v.z = v.z >= 0.f ? v.z : 0.2f * v.z;
        v.w = v.w >= 0.f ? v.w : 0.2f * v.w;
        out[i] = v;
    }
}

extern "C" void kernel_launch(void* const* d_in, const int* in_sizes, int n_in,
                              void* d_out, int out_size, void* d_ws, size_t ws_size,
                              hipStream_t stream) {
    const float* inputs  = (const float*)d_in[0];  // [NN, INCH]
    const float* w       = (const float*)d_in[1];  // [INCH, OUTCH]
    const float* b       = (const float*)d_in[2];  // [OUTCH]
    const float* a_att   = (const float*)d_in[3];  // [NREL]
    const float* adj_val = (const float*)d_in[4];  // [NREL, NEDGE]
    const int*   row     = (const int*)d_in[5];    // [NREL, NEDGE]
    const int*   col     = (const int*)d_in[6];    // [NREL, NEDGE]
    float*       out     = (float*)d_out;          // [NN, OUTCH]

    float* support = (float*)d_ws;                       // NN*OUTCH floats (51.2 MB)
    float* att     = support + (size_t)NN * OUTCH;       // NREL floats

    const int n4 = NN * OUTCH / 4;  // 3,200,000 float4s

    gc_zero_kernel<<<(n4 + 255) / 256, 256, 0, stream>>>((float4*)out, n4);
    gc_softmax_kernel<<<1, 32, 0, stream>>>(a_att, att);
    gc_gemm_wmma_kernel<<<NN / 16, 256, 0, stream>>>(inputs, w, b, support);

    dim3 sgrid(NEDGE / 8, NREL);
    gc_scatter_kernel<<<sgrid, 256, 0, stream>>>(support, att, adj_val, row, col, out);

    gc_leaky_kernel<<<(n4 + 255) / 256, 256, 0, stream>>>((float4*)out, n4);
}